// GAT_63024350101541
// MI455X (gfx1250) — compile-verified
//
#include <hip/hip_runtime.h>
#include <stdint.h>

// Problem constants (match reference)
#define NNODES 50000
#define NEDGES 800000
#define INDIM  32
#define HEADS  8
#define DHEAD  32
#define HD     256          // HEADS*DHEAD
#define NEG_SLOPE 0.2f

typedef __attribute__((ext_vector_type(2))) float v2f;
typedef __attribute__((ext_vector_type(8))) float v8f;

// ---------------------------------------------------------------------------
// WMMA fp32 GEMM:  Y[nrows,256] = X[nrows,K] * W[K,256]
// One wave computes one 16x16 output tile; K consumed 4 at a time via
// V_WMMA_F32_16X16X4_F32 (full fp32 precision, matches reference).
// A 16x4 layout : lane&15 -> M row, lane>>4 -> K-half, float2 = contiguous K
// B 4x16 layout : lane&15 -> N col, lane>>4 -> K-half, strided loads of W
// C/D 16x16     : lane&15 -> N col, VGPR v -> M = v + 8*(lane>>4)
// ---------------------------------------------------------------------------
__global__ __launch_bounds__(256) void gemm_wmma_f32(
    const float* __restrict__ X, const float* __restrict__ W,
    float* __restrict__ Y, int nrows, int K)
{
  const int lane  = threadIdx.x & 31;
  const int wgid  = blockIdx.x * 8 + (threadIdx.x >> 5);
  const int nColT = HD / 16;                 // 16 column tiles
  const int rowT  = wgid / nColT;
  const int colT  = wgid % nColT;
  if (rowT * 16 >= nrows) return;            // wave-uniform -> EXEC stays all-1s

  const int n    = lane & 15;                // A row index == C column index
  const int half = lane >> 4;                // selects K pair {0,1} vs {2,3}

  const float* Arow  = X + (size_t)(rowT * 16 + n) * K;
  const float* Bbase = W + colT * 16 + n;

  v8f c = {};
  for (int k0 = 0; k0 < K; k0 += 4) {
    v2f a, b;
    a.x = Arow[k0 + 2 * half];
    a.y = Arow[k0 + 2 * half + 1];
    b.x = Bbase[(size_t)(k0 + 2 * half)     * HD];
    b.y = Bbase[(size_t)(k0 + 2 * half + 1) * HD];
    c = __builtin_amdgcn_wmma_f32_16x16x4_f32(false, a, false, b,
                                              (short)0, c, false, false);
  }

  float* Yt = Y + (size_t)(rowT * 16) * HD + colT * 16 + n;
  #pragma unroll
  for (int v = 0; v < 8; ++v)
    Yt[(size_t)(v + 8 * half) * HD] = c[v];
}

// ---------------------------------------------------------------------------
// Zero per-layer accumulators (vectorized b128 stores):
// agg[N*256], encoded-max + sum [N*8]
// ---------------------------------------------------------------------------
__global__ void init_layer(uint4* __restrict__ menc,
                           float4* __restrict__ ssum,
                           float4* __restrict__ agg)
{
  int i = blockIdx.x * blockDim.x + threadIdx.x;
  const float4 z4 = make_float4(0.f, 0.f, 0.f, 0.f);
  if (i < NNODES * HD / 4) agg[i] = z4;
  if (i < NNODES * HEADS / 4) { menc[i] = make_uint4(0u, 0u, 0u, 0u); ssum[i] = z4; }
}

// ---------------------------------------------------------------------------
// el[n,h] = <feat[n,h,:], al[h,:]>, er likewise
// ---------------------------------------------------------------------------
__global__ void attn_scores(const float* __restrict__ feat,
                            const float* __restrict__ al,
                            const float* __restrict__ ar,
                            float* __restrict__ el, float* __restrict__ er)
{
  int i = blockIdx.x * blockDim.x + threadIdx.x;      // n*8 + h
  if (i >= NNODES * HEADS) return;
  int hh = i & 7;
  const float* f = feat + (size_t)(i >> 3) * HD + hh * DHEAD;
  const float* a = al + hh * DHEAD;
  const float* b = ar + hh * DHEAD;
  float sl = 0.f, sr = 0.f;
  #pragma unroll
  for (int d = 0; d < DHEAD; ++d) { float v = f[d]; sl += v * a[d]; sr += v * b[d]; }
  el[i] = sl; er[i] = sr;
}

// Monotone float <-> uint encoding so unsigned atomicMax == float max
__device__ __forceinline__ unsigned enc_f32(float f) {
  unsigned u = __float_as_uint(f);
  return (u & 0x80000000u) ? ~u : (u | 0x80000000u);
}
__device__ __forceinline__ float dec_f32(unsigned u) {
  unsigned b = (u & 0x80000000u) ? (u ^ 0x80000000u) : ~u;
  return __uint_as_float(b);
}

__device__ __forceinline__ float leaky(float v) {
  return v > 0.f ? v : NEG_SLOPE * v;
}

// ---------------------------------------------------------------------------
// Pass 1: segment max.  One thread per EDGE: vector-load all 8 heads of
// el[src] / er[dst] (4 x b128), 8 order-encoded atomicMax scatters.
// ---------------------------------------------------------------------------
__global__ void edge_max(const int* __restrict__ src, const int* __restrict__ dst,
                         const float* __restrict__ el, const float* __restrict__ er,
                         unsigned* __restrict__ menc)
{
  int e = blockIdx.x * blockDim.x + threadIdx.x;
  if (e >= NEDGES) return;
  const int s = src[e], d = dst[e];
  float4 l0 = *(const float4*)(el + (size_t)s * HEADS);
  float4 l1 = *(const float4*)(el + (size_t)s * HEADS + 4);
  float4 r0 = *(const float4*)(er + (size_t)d * HEADS);
  float4 r1 = *(const float4*)(er + (size_t)d * HEADS + 4);
  unsigned* md = menc + (size_t)d * HEADS;
  atomicMax(md + 0, enc_f32(leaky(l0.x + r0.x)));
  atomicMax(md + 1, enc_f32(leaky(l0.y + r0.y)));
  atomicMax(md + 2, enc_f32(leaky(l0.z + r0.z)));
  atomicMax(md + 3, enc_f32(leaky(l0.w + r0.w)));
  atomicMax(md + 4, enc_f32(leaky(l1.x + r1.x)));
  atomicMax(md + 5, enc_f32(leaky(l1.y + r1.y)));
  atomicMax(md + 6, enc_f32(leaky(l1.z + r1.z)));
  atomicMax(md + 7, enc_f32(leaky(l1.w + r1.w)));
}

// ---------------------------------------------------------------------------
// Pass 2: ex = exp(e - m[dst]); s[dst] += ex.  One thread per EDGE,
// coalesced 2 x b128 stores of ex (32B-aligned: e*8 floats).
// ---------------------------------------------------------------------------
__global__ void edge_exp(const int* __restrict__ src, const int* __restrict__ dst,
                         const float* __restrict__ el, const float* __restrict__ er,
                         const unsigned* __restrict__ menc,
                         float* __restrict__ ssum, float* __restrict__ ex)
{
  int e = blockIdx.x * blockDim.x + threadIdx.x;
  if (e >= NEDGES) return;
  const int s = src[e], d = dst[e];
  float4 l0 = *(const float4*)(el + (size_t)s * HEADS);
  float4 l1 = *(const float4*)(el + (size_t)s * HEADS + 4);
  float4 r0 = *(const float4*)(er + (size_t)d * HEADS);
  float4 r1 = *(const float4*)(er + (size_t)d * HEADS + 4);
  uint4 m0 = *(const uint4*)(menc + (size_t)d * HEADS);
  uint4 m1 = *(const uint4*)(menc + (size_t)d * HEADS + 4);
  float4 x0, x1;
  x0.x = __expf(leaky(l0.x + r0.x) - dec_f32(m0.x));
  x0.y = __expf(leaky(l0.y + r0.y) - dec_f32(m0.y));
  x0.z = __expf(leaky(l0.z + r0.z) - dec_f32(m0.z));
  x0.w = __expf(leaky(l0.w + r0.w) - dec_f32(m0.w));
  x1.x = __expf(leaky(l1.x + r1.x) - dec_f32(m1.x));
  x1.y = __expf(leaky(l1.y + r1.y) - dec_f32(m1.y));
  x1.z = __expf(leaky(l1.z + r1.z) - dec_f32(m1.z));
  x1.w = __expf(leaky(l1.w + r1.w) - dec_f32(m1.w));
  *(float4*)(ex + (size_t)e * HEADS)     = x0;
  *(float4*)(ex + (size_t)e * HEADS + 4) = x1;
  float* sd = ssum + (size_t)d * HEADS;
  atomicAdd(sd + 0, x0.x); atomicAdd(sd + 1, x0.y);
  atomicAdd(sd + 2, x0.z); atomicAdd(sd + 3, x0.w);
  atomicAdd(sd + 4, x1.x); atomicAdd(sd + 5, x1.y);
  atomicAdd(sd + 6, x1.z); atomicAdd(sd + 7, x1.w);
}

// ---------------------------------------------------------------------------
// Pass 3: agg[dst] += feat[src] * alpha.  4 edges per 256-thread block;
// 64 threads per edge, each covering 4 contiguous floats: one b128 gather
// from the L2-resident feat table + 4 f32 atomic scatters.
// Layer 1 also emits alpha to the output buffer.
// ---------------------------------------------------------------------------
template <bool WRITE_ALPHA>
__global__ __launch_bounds__(256) void edge_aggregate(
    const int* __restrict__ src, const int* __restrict__ dst,
    const float* __restrict__ feat, const float* __restrict__ ex,
    const float* __restrict__ ssum, float* __restrict__ agg,
    float* __restrict__ alphaOut)
{
  const int e  = blockIdx.x * 4 + (threadIdx.x >> 6); // edge for this 64-group
  const int g  = threadIdx.x & 63;                    // position in group
  const int j4 = g * 4;                               // first of 4 floats
  const int hh = g >> 3;                              // head = j4/32
  const int s = src[e], d = dst[e];
  const float alpha = ex[(size_t)e * HEADS + hh] / ssum[(size_t)d * HEADS + hh];
  float4 f = *(const float4*)(feat + (size_t)s * HD + j4);
  float* a = agg + (size_t)d * HD + j4;
  atomicAdd(a + 0, f.x * alpha);
  atomicAdd(a + 1, f.y * alpha);
  atomicAdd(a + 2, f.z * alpha);
  atomicAdd(a + 3, f.w * alpha);
  if (WRITE_ALPHA && g < HEADS)
    alphaOut[(size_t)e * HEADS + g] =
        ex[(size_t)e * HEADS + g] / ssum[(size_t)d * HEADS + g];
}

// h1 = relu(agg + b)  (in place, between layers; vectorized)
__global__ void bias_relu(float4* __restrict__ x, const float4* __restrict__ b)
{
  int i = blockIdx.x * blockDim.x + threadIdx.x;
  if (i >= NNODES * HD / 4) return;
  float4 v = x[i];
  float4 bb = b[i & (HD / 4 - 1)];
  v.x = fmaxf(v.x + bb.x, 0.f);
  v.y = fmaxf(v.y + bb.y, 0.f);
  v.z = fmaxf(v.z + bb.z, 0.f);
  v.w = fmaxf(v.w + bb.w, 0.f);
  x[i] = v;
}

// out[n,d] = mean_h(agg1[n,h,d] + b1[h,d])
__global__ void final_mean(const float* __restrict__ agg,
                           const float* __restrict__ b,
                           float* __restrict__ out)
{
  int i = blockIdx.x * blockDim.x + threadIdx.x;      // n*32 + d
  if (i >= NNODES * DHEAD) return;
  int n = i >> 5, d = i & 31;
  float s = 0.f;
  #pragma unroll
  for (int hh = 0; hh < HEADS; ++hh)
    s += agg[(size_t)n * HD + hh * DHEAD + d] + b[hh * DHEAD + d];
  out[i] = s * (1.0f / HEADS);
}

// ---------------------------------------------------------------------------
extern "C" void kernel_launch(void* const* d_in, const int* in_sizes, int n_in,
                              void* d_out, int out_size, void* d_ws, size_t ws_size,
                              hipStream_t stream)
{
  const float* h   = (const float*)d_in[0];
  const int*   src = (const int*)  d_in[1];
  const int*   dst = (const int*)  d_in[2];
  const float* W0  = (const float*)d_in[3];
  const float* al0 = (const float*)d_in[4];
  const float* ar0 = (const float*)d_in[5];
  const float* b0  = (const float*)d_in[6];
  const float* W1  = (const float*)d_in[7];
  const float* al1 = (const float*)d_in[8];
  const float* ar1 = (const float*)d_in[9];
  const float* b1  = (const float*)d_in[10];

  // Workspace carving (all fp32; every offset 16B-aligned)
  char* ws = (char*)d_ws;
  const size_t szNF = (size_t)NNODES * HD * sizeof(float);      // 51.2 MB
  const size_t szNH = (size_t)NNODES * HEADS * sizeof(float);   // 1.6 MB
  float*    feat = (float*)(ws);                       // projected features
  float*    hbuf = (float*)(ws + szNF);                // agg0 -> h1 (in place)
  float*    agg1 = (float*)(ws + 2 * szNF);
  float*    el   = (float*)(ws + 3 * szNF);
  float*    er   = (float*)(ws + 3 * szNF + szNH);
  unsigned* menc = (unsigned*)(ws + 3 * szNF + 2 * szNH);
  float*    ssum = (float*)(ws + 3 * szNF + 3 * szNH);
  float*    ex   = (float*)(ws + 3 * szNF + 4 * szNH);
  (void)in_sizes; (void)n_in; (void)out_size; (void)ws_size;

  float* out      = (float*)d_out;                     // [N,32] head-mean
  float* alphaOut = out + (size_t)NNODES * DHEAD;      // [E,8]  layer-1 alpha

  const int TB = 256;
  const int gNF4 = (NNODES * HD / 4 + TB - 1) / TB;    // float4-based grids
  const int gNH  = (NNODES * HEADS + TB - 1) / TB;
  const int gE   = (NEDGES + TB - 1) / TB;             // thread-per-edge grids
  const int gAgg = NEDGES / 4;                         // 4 edges per block
  const int nTiles = ((NNODES + 15) / 16) * (HD / 16); // waves needed
  const int gGemm = (nTiles + 7) / 8;                  // 8 waves per block

  // ----- Layer 0 -----
  init_layer<<<gNF4, TB, 0, stream>>>((uint4*)menc, (float4*)ssum, (float4*)hbuf);
  gemm_wmma_f32<<<gGemm, TB, 0, stream>>>(h, W0, feat, NNODES, INDIM);
  attn_scores<<<gNH, TB, 0, stream>>>(feat, al0, ar0, el, er);
  edge_max<<<gE, TB, 0, stream>>>(src, dst, el, er, menc);
  edge_exp<<<gE, TB, 0, stream>>>(src, dst, el, er, menc, ssum, ex);
  edge_aggregate<false><<<gAgg, TB, 0, stream>>>(src, dst, feat, ex, ssum,
                                                 hbuf, nullptr);
  bias_relu<<<gNF4, TB, 0, stream>>>((float4*)hbuf, (const float4*)b0);

  // ----- Layer 1 -----
  init_layer<<<gNF4, TB, 0, stream>>>((uint4*)menc, (float4*)ssum, (float4*)agg1);
  gemm_wmma_f32<<<gGemm, TB, 0, stream>>>(hbuf, W1, feat, NNODES, HD);
  attn_scores<<<gNH, TB, 0, stream>>>(feat, al1, ar1, el, er);
  edge_max<<<gE, TB, 0, stream>>>(src, dst, el, er, menc);
  edge_exp<<<gE, TB, 0, stream>>>(src, dst, el, er, menc, ssum, ex);
  edge_aggregate<true><<<gAgg, TB, 0, stream>>>(src, dst, feat, ex, ssum,
                                                agg1, alphaOut);
  final_mean<<<(NNODES * DHEAD + TB - 1) / TB, TB, 0, stream>>>(agg1, b1, out);
}